// TinyMemory_33139967656581
// MI455X (gfx1250) — compile-verified
//
#include <hip/hip_runtime.h>

typedef __attribute__((ext_vector_type(2))) float v2f;
typedef __attribute__((ext_vector_type(4))) float v4f;
typedef __attribute__((ext_vector_type(8))) float v8f;

#define B_DIM 256
#define M_DIM 256
#define C_DIM 1024
#define ALPHA 0.1f
#define SCALE 0.03125f  /* 1/sqrt(1024) */

// ---------------------------------------------------------------------------
// Kernel A: sims[b,m] = dot(in[b,:], mm[m,:])   (16x16 tile per wave, K=1024)
// A-matrix layout (32-bit 16x4): lane<16 holds K={0,1}, lane>=16 holds K={2,3}
// for row M = lane&15. B mirrors with N = lane&15. D: VGPR r -> M=r+8*(lane>=16).
// ---------------------------------------------------------------------------
__global__ __launch_bounds__(128) void sims_gemm(const float* __restrict__ in,
                                                 const float* __restrict__ mm,
                                                 float* __restrict__ sims) {
  int wave = threadIdx.x >> 5;
  int lane = threadIdx.x & 31;
  int tile = blockIdx.x * 4 + wave;          // 256 tiles total
  int tr = (tile >> 4) << 4;                 // b tile base
  int tc = (tile & 15) << 4;                 // m tile base
  int half = lane >> 4;
  int i16 = lane & 15;
  const float* ap = in + (size_t)(tr + i16) * C_DIM + half * 2;
  const float* bp = mm + (size_t)(tc + i16) * C_DIM + half * 2;
  v8f acc = {};
#pragma unroll 8
  for (int k = 0; k < C_DIM; k += 4) {
    v2f a = *(const v2f*)(ap + k);
    v2f b = *(const v2f*)(bp + k);
    acc = __builtin_amdgcn_wmma_f32_16x16x4_f32(false, a, false, b,
                                                (short)0, acc, false, false);
  }
  int col = tc + i16;
#pragma unroll
  for (int r = 0; r < 8; ++r) {
    sims[(tr + r + 8 * half) * M_DIM + col] = acc[r];
  }
}

// ---------------------------------------------------------------------------
// Kernel B: per row b -> argmax ci, norm2, dkl_write, softmax over fixed scores
// ---------------------------------------------------------------------------
__global__ __launch_bounds__(256) void rowstats(const float* __restrict__ in,
                                                const float* __restrict__ mm,
                                                const float* __restrict__ sims,
                                                float* __restrict__ attn,
                                                int* __restrict__ ci_out,
                                                float* __restrict__ aci_out,
                                                float* __restrict__ dklw_out) {
  __shared__ float sv[256];
  __shared__ int si[256];
  int b = blockIdx.x;
  int t = threadIdx.x;

  // ---- argmax over sims[b,:] (first-max tie break like jnp.argmax) ----
  float v = sims[b * M_DIM + t];
  sv[t] = v; si[t] = t;
  __syncthreads();
  for (int s = 128; s > 0; s >>= 1) {
    if (t < s) {
      float v2 = sv[t + s]; int i2 = si[t + s];
      if (v2 > sv[t] || (v2 == sv[t] && i2 < si[t])) { sv[t] = v2; si[t] = i2; }
    }
    __syncthreads();
  }
  int ci = si[0];
  float simci = sv[0];
  __syncthreads();

  // ---- norm2 = ||in[b]||^2 ----
  float acc = 0.f;
  for (int c = t; c < C_DIM; c += 256) {
    float x = in[b * C_DIM + c];
    acc += x * x;
  }
  sv[t] = acc; __syncthreads();
  for (int s = 128; s > 0; s >>= 1) { if (t < s) sv[t] += sv[t + s]; __syncthreads(); }
  float norm2 = sv[0];
  __syncthreads();

  // ---- dkl_write = 0.5*||in[b]-mm[ci]||^2 ----
  acc = 0.f;
  for (int c = t; c < C_DIM; c += 256) {
    float d = in[b * C_DIM + c] - mm[ci * C_DIM + c];
    acc += d * d;
  }
  sv[t] = acc; __syncthreads();
  for (int s = 128; s > 0; s >>= 1) { if (t < s) sv[t] += sv[t + s]; __syncthreads(); }
  float dklw = 0.5f * sv[0];
  __syncthreads();

  // ---- softmax over scores (sims with the ci entry replaced) ----
  float sc = (t == ci) ? ((1.0f - ALPHA) * simci + ALPHA * norm2) * SCALE
                       : sims[b * M_DIM + t] * SCALE;
  sv[t] = sc; __syncthreads();
  for (int s = 128; s > 0; s >>= 1) { if (t < s) sv[t] = fmaxf(sv[t], sv[t + s]); __syncthreads(); }
  float mx = sv[0];
  __syncthreads();
  float e = expf(sc - mx);
  sv[t] = e; __syncthreads();
  for (int s = 128; s > 0; s >>= 1) { if (t < s) sv[t] += sv[t + s]; __syncthreads(); }
  float a = e / sv[0];
  attn[b * M_DIM + t] = a;
  if (t == ci) aci_out[b] = a;
  if (t == 0) { ci_out[b] = ci; dklw_out[b] = dklw; }
}

// ---------------------------------------------------------------------------
// Kernel C: z_read_raw = attn @ mm   [256 x 1024], K = 256 (WMMA f32)
// ---------------------------------------------------------------------------
__global__ __launch_bounds__(128) void zread_gemm(const float* __restrict__ attn,
                                                  const float* __restrict__ mm,
                                                  float* __restrict__ zr) {
  int wave = threadIdx.x >> 5;
  int lane = threadIdx.x & 31;
  int tile = blockIdx.x * 4 + wave;          // 1024 tiles: 16 row x 64 col
  int tr = (tile >> 6) << 4;                 // b tile base
  int tc = (tile & 63) << 4;                 // c tile base
  int half = lane >> 4;
  int i16 = lane & 15;
  const float* ap = attn + (tr + i16) * M_DIM + half * 2;
  const float* bp = mm + (size_t)(half * 2) * C_DIM + tc + i16;
  v8f acc = {};
#pragma unroll 4
  for (int k = 0; k < M_DIM; k += 4) {
    v2f a = *(const v2f*)(ap + k);
    v2f b;
    b.x = bp[(size_t)k * C_DIM];
    b.y = bp[(size_t)k * C_DIM + C_DIM];
    acc = __builtin_amdgcn_wmma_f32_16x16x4_f32(false, a, false, b,
                                                (short)0, acc, false, false);
  }
  int col = tc + i16;
#pragma unroll
  for (int r = 0; r < 8; ++r) {
    zr[(size_t)(tr + r + 8 * half) * C_DIM + col] = acc[r];
  }
}

// ---------------------------------------------------------------------------
// Kernel D: rank-1 correction of z_read + kl_div = dkl_write + dkl_read
//   z_read += 0.1 * attn[b,ci] * (in[b] - mm[ci])
// ---------------------------------------------------------------------------
__global__ __launch_bounds__(256) void epilogue(const float* __restrict__ in,
                                                const float* __restrict__ mm,
                                                const int* __restrict__ ci_arr,
                                                const float* __restrict__ aci_arr,
                                                const float* __restrict__ dklw_arr,
                                                float* __restrict__ zr,
                                                float* __restrict__ kl) {
  __shared__ float sv[256];
  int b = blockIdx.x;
  int t = threadIdx.x;
  int ci = ci_arr[b];
  float aci = aci_arr[b];
  float acc = 0.f;
  for (int c = t; c < C_DIM; c += 256) {
    float x = in[b * C_DIM + c];
    float z = zr[(size_t)b * C_DIM + c] + ALPHA * aci * (x - mm[ci * C_DIM + c]);
    zr[(size_t)b * C_DIM + c] = z;
    float d = z - x;
    acc += d * d;
  }
  sv[t] = acc; __syncthreads();
  for (int s = 128; s > 0; s >>= 1) { if (t < s) sv[t] += sv[t + s]; __syncthreads(); }
  if (t == 0) kl[b] = dklw_arr[b] + 0.5f * sv[0];
}

// ---------------------------------------------------------------------------
// Kernel E (runs LAST, overwrites the um region incl. the sims/attn scratch):
//   updated_memory[b,m,:] = (m==ci[b]) ? 0.9*mm[ci]+0.1*in[b] : mm[m]
// 256 MB streamed with non-temporal stores (output > L2 capacity).
// Uses native ext-vector v4f (clang nontemporal builtin rejects HIP float4).
// ---------------------------------------------------------------------------
__global__ __launch_bounds__(256) void write_mem(const float* __restrict__ in,
                                                 const float* __restrict__ mm,
                                                 const int* __restrict__ ci_arr,
                                                 float* __restrict__ um) {
  int m = blockIdx.x;
  int b = blockIdx.y;
  int t = threadIdx.x;
  const v4f* src = (const v4f*)(mm + (size_t)m * C_DIM);
  v4f v = src[t];
  if (m == ci_arr[b]) {
    const v4f* iv = (const v4f*)(in + (size_t)b * C_DIM);
    v4f w = iv[t];
    v = (1.0f - ALPHA) * v + ALPHA * w;
  }
  v4f* dst = (v4f*)(um + ((size_t)b * M_DIM + m) * C_DIM);
  __builtin_nontemporal_store(v, dst + t);
}

// ---------------------------------------------------------------------------
extern "C" void kernel_launch(void* const* d_in, const int* in_sizes, int n_in,
                              void* d_out, int out_size, void* d_ws, size_t ws_size,
                              hipStream_t stream) {
  const float* in = (const float*)d_in[0];   // [256,1024] f32
  const float* mm = (const float*)d_in[1];   // [256,1024] f32
  float* out = (float*)d_out;
  float* zr = out;                                      // [256,1024]
  float* um = out + (size_t)B_DIM * C_DIM;              // [256,256,1024]
  float* kl = um + (size_t)B_DIM * M_DIM * C_DIM;       // [256]

  // Large scratch lives at the front of the updated_memory output region;
  // write_mem (which overwrites it) runs last. Tiny arrays (3 KB) in d_ws.
  float* sims = um;                       // 256*256 f32
  float* attn = um + B_DIM * M_DIM;       // 256*256 f32
  int*   ci   = (int*)d_ws;               // 256 i32
  float* aci  = (float*)d_ws + 256;       // 256 f32
  float* dklw = (float*)d_ws + 512;       // 256 f32

  sims_gemm <<<64, 128, 0, stream>>>(in, mm, sims);
  rowstats  <<<B_DIM, 256, 0, stream>>>(in, mm, sims, attn, ci, aci, dklw);
  zread_gemm<<<256, 128, 0, stream>>>(attn, mm, zr);
  epilogue  <<<B_DIM, 256, 0, stream>>>(in, mm, ci, aci, dklw, zr, kl);
  write_mem <<<dim3(M_DIM, B_DIM), 256, 0, stream>>>(in, mm, ci, um);
}